// AttnDecoderRNN_46127948759381
// MI455X (gfx1250) — compile-verified
//
#include <hip/hip_runtime.h>
#include <hip/hip_bf16.h>

#define T_ 16
#define B_ 32
#define S_ 256
#define H_ 256
#define E_ 300
#define V_ 1000
#define G4H (4 * H_)

typedef float v2f __attribute__((ext_vector_type(2)));
typedef float v8f __attribute__((ext_vector_type(8)));

__device__ __forceinline__ float sigf(float x) { return 1.0f / (1.0f + __expf(-x)); }

// ---------------------------------------------------------------------------
// Generic fp32 WMMA GEMM: C[M,N] = A[M,K] * W[N,K]^T + bias[N]
// One wave (32 lanes) computes one 16x16 C tile via V_WMMA_F32_16X16X4_F32.
// blockDim = (32, 4): 4 independent N-tiles per block.
// A layout (16x4 f32): lane%16 = M row, lane/16 selects K pair {0,1}|{2,3}.
// B layout mirrors with lane%16 = N column. C: v8f, M = v + 8*(lane>=16).
// GUARD=true only for the logits GEMM (N=1000, tile 62 partial); the exact-N
// instantiation has a mask-free inner loop (pure load + v_wmma).
// ---------------------------------------------------------------------------
template <bool GUARD>
__global__ __launch_bounds__(128) void wmma_gemm_bias(
    const float* __restrict__ A, const float* __restrict__ W,
    const float* __restrict__ bias, float* __restrict__ C, int K, int N) {
  const int lane = threadIdx.x;   // 0..31
  const int row  = lane & 15;
  const int hi   = lane >> 4;     // 0 or 1
  const int m0   = blockIdx.x * 16;
  const int n0   = (blockIdx.y * blockDim.y + threadIdx.y) * 16;
  if (GUARD && n0 >= N) return;            // wave-uniform (y is per-wave)
  const int  n     = n0 + row;
  const bool colOK = GUARD ? (n < N) : true;
  const int  ncl   = colOK ? n : 0;

  const float* Ap = A + (size_t)(m0 + row) * K + 2 * hi;
  const float* Wp = W + (size_t)ncl * K + 2 * hi;

  v8f acc = {};
  for (int k = 0; k < K; k += 4) {
    float2 av = *(const float2*)(Ap + k);
    float2 bv = *(const float2*)(Wp + k);
    if (GUARD && !colOK) { bv.x = 0.0f; bv.y = 0.0f; }
    v2f a; a.x = av.x; a.y = av.y;
    v2f b; b.x = bv.x; b.y = bv.y;
    acc = __builtin_amdgcn_wmma_f32_16x16x4_f32(false, a, false, b,
                                                (short)0, acc, false, false);
  }
  if (colOK) {
    const float bb = bias[n];
#pragma unroll
    for (int v = 0; v < 8; ++v) {
      const int m = m0 + v + 8 * hi;
      C[(size_t)m * N + n] = acc[v] + bb;
    }
  }
}

// ---------------------------------------------------------------------------
// One LSTM time step. One wave per (16 x 16) (B x H) tile; computes all four
// gate tiles (i,f,g,o) against W_hh with a shared A operand, then applies the
// cell update in registers. grid = (B/16, H/16), blockDim = 32.
// ---------------------------------------------------------------------------
__global__ __launch_bounds__(32) void lstm_step_kernel(
    const float* __restrict__ hprev,  // [B,H]
    const float* __restrict__ Whh,    // [4H,H]
    const float* __restrict__ bhh,    // [4H]
    const float* __restrict__ xg,     // [B,4H]  (x @ W_ih^T + b_ih) for this t
    float* __restrict__ cbuf,         // [B,H] in/out
    float* __restrict__ hout,         // [B,H]
    float* __restrict__ hT, float* __restrict__ cT, int is_last) {
  const int lane = threadIdx.x;
  const int row  = lane & 15;
  const int hi   = lane >> 4;
  const int m0   = blockIdx.x * 16;   // batch
  const int n0   = blockIdx.y * 16;   // hidden
  const int n    = n0 + row;

  const float* Ap = hprev + (size_t)(m0 + row) * H_ + 2 * hi;
  const float* W0 = Whh + (size_t)(0 * H_ + n) * H_ + 2 * hi;
  const float* W1 = Whh + (size_t)(1 * H_ + n) * H_ + 2 * hi;
  const float* W2 = Whh + (size_t)(2 * H_ + n) * H_ + 2 * hi;
  const float* W3 = Whh + (size_t)(3 * H_ + n) * H_ + 2 * hi;

  v8f ci = {}, cf = {}, cg = {}, co = {};
  for (int k = 0; k < H_; k += 4) {
    float2 av = *(const float2*)(Ap + k);
    v2f a; a.x = av.x; a.y = av.y;
    float2 b0 = *(const float2*)(W0 + k); v2f v0; v0.x = b0.x; v0.y = b0.y;
    float2 b1 = *(const float2*)(W1 + k); v2f v1; v1.x = b1.x; v1.y = b1.y;
    float2 b2 = *(const float2*)(W2 + k); v2f v2; v2.x = b2.x; v2.y = b2.y;
    float2 b3 = *(const float2*)(W3 + k); v2f v3; v3.x = b3.x; v3.y = b3.y;
    ci = __builtin_amdgcn_wmma_f32_16x16x4_f32(false, a, false, v0, (short)0, ci, false, false);
    cf = __builtin_amdgcn_wmma_f32_16x16x4_f32(false, a, false, v1, (short)0, cf, false, false);
    cg = __builtin_amdgcn_wmma_f32_16x16x4_f32(false, a, false, v2, (short)0, cg, false, false);
    co = __builtin_amdgcn_wmma_f32_16x16x4_f32(false, a, false, v3, (short)0, co, false, false);
  }
  const float bi = bhh[n];
  const float bf = bhh[H_ + n];
  const float bg = bhh[2 * H_ + n];
  const float bo = bhh[3 * H_ + n];
#pragma unroll
  for (int v = 0; v < 8; ++v) {
    const int m = m0 + v + 8 * hi;
    const size_t gb = (size_t)m * G4H;
    const float gi = ci[v] + xg[gb + n]            + bi;
    const float gf = cf[v] + xg[gb + H_ + n]       + bf;
    const float gg = cg[v] + xg[gb + 2 * H_ + n]   + bg;
    const float go = co[v] + xg[gb + 3 * H_ + n]   + bo;
    const float cold = cbuf[(size_t)m * H_ + n];
    const float cnew = sigf(gf) * cold + sigf(gi) * tanhf(gg);
    const float hnew = sigf(go) * tanhf(cnew);
    cbuf[(size_t)m * H_ + n] = cnew;
    hout[(size_t)m * H_ + n] = hnew;
    if (is_last) {
      hT[(size_t)m * H_ + n] = hnew;
      cT[(size_t)m * H_ + n] = cnew;
    }
  }
}

// ---------------------------------------------------------------------------
// Fused additive-attention for one (t,b): score(s) = wa . tanh(enc_t + dec_t)
// + ba, mask s>=len, softmax over S in LDS, then context = sum_s p * enc.
// 256 threads = 8 waves; each wave computes 32 scores with a wave32 reduce.
// Also assembles combined[t,b, :] = [context, h_t] for the logits GEMM.
// ---------------------------------------------------------------------------
__global__ __launch_bounds__(256) void attn_kernel(
    const float* __restrict__ enc_t,   // [S,B,H]
    const float* __restrict__ dec_t,   // [T*B,H]
    const float* __restrict__ enc,     // encoder_outputs [S,B,H]
    const float* __restrict__ outs,    // [T,B,H]
    const float* __restrict__ wa, const float* __restrict__ ba,
    const int* __restrict__ lens,
    float* __restrict__ ctx_out,       // [B,T,H]
    float* __restrict__ combined) {    // [T*B,2H]
  __shared__ float sc[S_];
  __shared__ float red[256];
  const int tid  = threadIdx.x;
  const int lane = tid & 31;
  const int wv   = tid >> 5;
  const int tb   = blockIdx.x;        // t*B + b
  const int t    = tb / B_;
  const int b    = tb % B_;

  float dreg[8], wreg[8];
#pragma unroll
  for (int j = 0; j < 8; ++j) {
    const int h = lane + 32 * j;
    dreg[j] = dec_t[(size_t)tb * H_ + h];
    wreg[j] = wa[h];
  }
  const float ba0 = ba[0];
  for (int jj = 0; jj < 32; ++jj) {
    const int s = wv * 32 + jj;
    const float* ep = enc_t + ((size_t)s * B_ + b) * H_;
    float acc = 0.0f;
#pragma unroll
    for (int j = 0; j < 8; ++j) {
      const int h = lane + 32 * j;
      acc += tanhf(ep[h] + dreg[j]) * wreg[j];
    }
#pragma unroll
    for (int off = 16; off > 0; off >>= 1) acc += __shfl_xor(acc, off, 32);
    if (lane == 0) sc[s] = acc + ba0;
  }
  __syncthreads();

  const int len = lens[b];
  const float v = (tid < len) ? sc[tid] : -__builtin_inff();
  red[tid] = v; __syncthreads();
  for (int st = 128; st > 0; st >>= 1) {
    if (tid < st) red[tid] = fmaxf(red[tid], red[tid + st]);
    __syncthreads();
  }
  const float mx = red[0]; __syncthreads();
  const float p = __expf(v - mx);          // masked lanes -> 0
  red[tid] = p; __syncthreads();
  for (int st = 128; st > 0; st >>= 1) {
    if (tid < st) red[tid] += red[tid + st];
    __syncthreads();
  }
  const float sum = red[0]; __syncthreads();
  sc[tid] = p / sum;
  __syncthreads();

  // context for h = tid (coalesced over the block)
  float ctx = 0.0f;
  for (int s = 0; s < S_; ++s)
    ctx += sc[s] * enc[((size_t)s * B_ + b) * H_ + tid];
  ctx_out[((size_t)b * T_ + t) * H_ + tid] = ctx;
  combined[(size_t)tb * 2 * H_ + tid]       = ctx;
  combined[(size_t)tb * 2 * H_ + H_ + tid]  = outs[(size_t)tb * H_ + tid];
}

// ---------------------------------------------------------------------------
// Softmax over V=1000 per row of logits [T*B, V].
// ---------------------------------------------------------------------------
__global__ __launch_bounds__(256) void softmaxV_kernel(
    const float* __restrict__ logits, float* __restrict__ prob) {
  __shared__ float red[256];
  const int r = blockIdx.x, tid = threadIdx.x;
  const float* lp = logits + (size_t)r * V_;
  float mx = -__builtin_inff();
  for (int v = tid; v < V_; v += 256) mx = fmaxf(mx, lp[v]);
  red[tid] = mx; __syncthreads();
  for (int st = 128; st > 0; st >>= 1) {
    if (tid < st) red[tid] = fmaxf(red[tid], red[tid + st]);
    __syncthreads();
  }
  mx = red[0]; __syncthreads();
  float sum = 0.0f;
  for (int v = tid; v < V_; v += 256) sum += __expf(lp[v] - mx);
  red[tid] = sum; __syncthreads();
  for (int st = 128; st > 0; st >>= 1) {
    if (tid < st) red[tid] += red[tid + st];
    __syncthreads();
  }
  sum = red[0];
  const float inv = 1.0f / sum;
  for (int v = tid; v < V_; v += 256)
    prob[(size_t)r * V_ + v] = __expf(lp[v] - mx) * inv;
}

// ---------------------------------------------------------------------------
__global__ void gather_emb(const int* __restrict__ idx,
                           const float* __restrict__ table,
                           float* __restrict__ emb) {
  const int i = blockIdx.x * blockDim.x + threadIdx.x;
  if (i >= T_ * B_ * E_) return;
  const int rb = i / E_;
  const int e  = i % E_;
  emb[i] = table[(size_t)idx[rb] * E_ + e];
}

// ---------------------------------------------------------------------------
extern "C" void kernel_launch(void* const* d_in, const int* in_sizes, int n_in,
                              void* d_out, int out_size, void* d_ws, size_t ws_size,
                              hipStream_t stream) {
  (void)in_sizes; (void)n_in; (void)out_size; (void)ws_size;
  const int*   tgt   = (const int*)d_in[0];
  const float* h0    = (const float*)d_in[1];
  const float* c0    = (const float*)d_in[2];
  const float* enc   = (const float*)d_in[3];
  const int*   lens  = (const int*)d_in[4];
  const float* table = (const float*)d_in[5];
  const float* W_ih  = (const float*)d_in[6];
  const float* W_hh  = (const float*)d_in[7];
  const float* b_ih  = (const float*)d_in[8];
  const float* b_hh  = (const float*)d_in[9];
  const float* We    = (const float*)d_in[10];
  const float* be    = (const float*)d_in[11];
  const float* Wd    = (const float*)d_in[12];
  const float* bd    = (const float*)d_in[13];
  const float* wa    = (const float*)d_in[14];
  const float* ba    = (const float*)d_in[15];
  const float* W_out = (const float*)d_in[16];
  const float* b_out = (const float*)d_in[17];

  float* out_prob = (float*)d_out;                       // [T,B,V]
  float* hT       = out_prob + (size_t)T_ * B_ * V_;     // [1,B,H]
  float* cT       = hT + (size_t)B_ * H_;                // [1,B,H]
  float* ctx_out  = cT + (size_t)B_ * H_;                // [B,T,H]

  float* ws     = (float*)d_ws;
  float* enc_t  = ws;  ws += (size_t)S_ * B_ * H_;       // [S,B,H]
  float* outs   = ws;  ws += (size_t)T_ * B_ * H_;       // [T,B,H]
  float* cbuf   = ws;  ws += (size_t)B_ * H_;            // [B,H]
  float* xg     = ws;  ws += (size_t)T_ * B_ * G4H;      // [T,B,4H]
  float* emb    = ws;  ws += (size_t)T_ * B_ * E_;       // [T,B,E]
  float* dect   = ws;  ws += (size_t)T_ * B_ * H_;       // [T,B,H]
  float* comb   = ws;  ws += (size_t)T_ * B_ * 2 * H_;   // [T*B,2H]
  float* logits = ws;  ws += (size_t)T_ * B_ * V_;       // [T*B,V]

  // 1) embedding gather
  { const int n = T_ * B_ * E_;
    gather_emb<<<(n + 255) / 256, 256, 0, stream>>>(tgt, table, emb); }

  // 2) x-projection: xg[T*B,4H] = emb[T*B,E] @ W_ih^T + b_ih  (exact N)
  { dim3 g(T_ * B_ / 16, (G4H / 16) / 4); dim3 blk(32, 4);
    wmma_gemm_bias<false><<<g, blk, 0, stream>>>(emb, W_ih, b_ih, xg, E_, G4H); }

  // 3) encoder projection: enc_t[S*B,H] = enc[S*B,H] @ We^T + be  (exact N)
  { dim3 g(S_ * B_ / 16, (H_ / 16) / 4); dim3 blk(32, 4);
    wmma_gemm_bias<false><<<g, blk, 0, stream>>>(enc, We, be, enc_t, H_, H_); }

  // 4) cell-state init (graph-capture-safe async d2d)
  hipMemcpyAsync(cbuf, c0, (size_t)B_ * H_ * sizeof(float),
                 hipMemcpyDeviceToDevice, stream);

  // 5) LSTM recurrence
  for (int t = 0; t < T_; ++t) {
    const float* hprev = (t == 0) ? h0 : (outs + (size_t)(t - 1) * B_ * H_);
    dim3 g(B_ / 16, H_ / 16);
    lstm_step_kernel<<<g, 32, 0, stream>>>(
        hprev, W_hh, b_hh, xg + (size_t)t * B_ * G4H,
        cbuf, outs + (size_t)t * B_ * H_, hT, cT, (t == T_ - 1) ? 1 : 0);
  }

  // 6) decoder projection: dect[T*B,H] = outs @ Wd^T + bd  (exact N)
  { dim3 g(T_ * B_ / 16, (H_ / 16) / 4); dim3 blk(32, 4);
    wmma_gemm_bias<false><<<g, blk, 0, stream>>>(outs, Wd, bd, dect, H_, H_); }

  // 7) fused attention: scores + mask + softmax + context (+ combined)
  attn_kernel<<<T_ * B_, 256, 0, stream>>>(enc_t, dect, enc, outs, wa, ba,
                                           lens, ctx_out, comb);

  // 8) logits: [T*B,1000] = comb[T*B,2H] @ W_out^T + b_out (N=1000 guarded)
  { const int ntiles = (V_ + 15) / 16;  // 63
    dim3 g(T_ * B_ / 16, (ntiles + 3) / 4); dim3 blk(32, 4);
    wmma_gemm_bias<true><<<g, blk, 0, stream>>>(comb, W_out, b_out, logits,
                                                2 * H_, V_); }

  // 9) softmax over V
  softmaxV_kernel<<<T_ * B_, 256, 0, stream>>>(logits, out_prob);
}